// GroupNorm2dInteger_17970143167184
// MI455X (gfx1250) — compile-verified
//
#include <hip/hip_runtime.h>
#include <hip/hip_bf16.h>
#include <stdint.h>

// ---------------------------------------------------------------------------
// GroupNorm2dInteger for MI455X (gfx1250), compile-only loop.
// x: (128, 64, 56, 56) fp32.  Per-batch reduction over N = 64*56*56 = 200704.
// Memory-bound (AI ~0.6 flop/byte): data movement via TDM tensor_load_to_lds
// (double-buffered, s_wait_tensorcnt), cross-lane reduction via
// v_wmma_f32_16x16x4_f32 (sum and sumsq folded in one WMMA).
// ---------------------------------------------------------------------------

#define NBATCH        128
#define N_PER_BATCH   (64 * 56 * 56)                 // 200704 = 49 * 4096
#define TOTAL_ELEMS   ((size_t)NBATCH * (size_t)N_PER_BATCH)
#define TILE          4096                           // floats per TDM tile (16 KB)
#define SPLIT         7                              // blocks per batch
#define TILES_PER_BLK (N_PER_BATCH / (SPLIT * TILE)) // 7
#define SQRT2_INV_Q15 23170                          // floor(2^15 / sqrt(2))

typedef unsigned int v4u __attribute__((ext_vector_type(4)));
typedef int          v8i __attribute__((ext_vector_type(8)));
typedef int          v4i __attribute__((ext_vector_type(4)));
typedef float        v2f __attribute__((ext_vector_type(2)));
typedef float        v8f __attribute__((ext_vector_type(8)));

#if defined(__has_include)
#if __has_include(<hip/amd_detail/amd_gfx1250_TDM.h>)
#define TDM_SIX_ARGS 1
#endif
#endif

#if defined(__HIP_DEVICE_COMPILE__)
#if __has_builtin(__builtin_amdgcn_tensor_load_to_lds)
#define HAVE_TDM 1
#endif
#if __has_builtin(__builtin_amdgcn_wmma_f32_16x16x4_f32)
#define HAVE_WMMA_F32X4 1
#endif
#endif

#if defined(HAVE_TDM)
// One TDM DMA of TILE contiguous fp32 from global into LDS.
// D# per CDNA5 ISA 10.8: group0 = count/lds_addr/global_addr/type=2;
// group1 = data_size=4B, tensor_dim0=TILE, tensor_dim1=1, tile_dim0=TILE,
// tile_dim1=1, tensor_dim0_stride=TILE.  Groups 2/3 unused (<=2D tensor).
__device__ __forceinline__ void tdm_load_tile(uint32_t lds_byte_off,
                                              const float* gsrc) {
  uint64_t ga = (uint64_t)(uintptr_t)gsrc;
  v4u g0 = { 1u,                                     // count = 1 (valid)
             lds_byte_off,                           // LDS dest byte address
             (uint32_t)ga,                           // global_addr[31:0]
             (uint32_t)((ga >> 32) & 0x01FFFFFFu) | (2u << 30) }; // [56:32]|type=2
  v8i g1 = { (int)(2u << 16),                        // data_size = 2 -> 4 bytes
             (int)((unsigned)(TILE & 0xFFFF) << 16), // tensor_dim0[15:0]
             (int)(((unsigned)TILE >> 16) | (1u << 16)), // dim0 hi | tensor_dim1=1
             (int)((unsigned)TILE << 16),            // tile_dim0 = TILE
             1,                                      // tile_dim1 = 1, tile_dim2 = 0
             TILE,                                   // tensor_dim0_stride lo
             0, 0 };
  v4i z4 = { 0, 0, 0, 0 };
#if defined(TDM_SIX_ARGS)
  v8i z8 = { 0, 0, 0, 0, 0, 0, 0, 0 };
  __builtin_amdgcn_tensor_load_to_lds(g0, g1, z4, z4, z8, 0);
#else
  __builtin_amdgcn_tensor_load_to_lds(g0, g1, z4, z4, 0);
#endif
}
#endif // HAVE_TDM

// ---------------------------------------------------------------------------
// Pass 0: zero the per-batch accumulators (sum, sumsq) in workspace.
// ---------------------------------------------------------------------------
__global__ void gni_zero(float* __restrict__ ws) {
  if (threadIdx.x < 2 * NBATCH) ws[threadIdx.x] = 0.0f;
}

// ---------------------------------------------------------------------------
// Pass 1: per-batch sum(x) and sum(x^2).  TDM double-buffered through LDS,
// wave reduction via one v_wmma_f32_16x16x4_f32 (both quantities at once).
// Grid: NBATCH * SPLIT blocks of 256 threads (8 wave32).
// ---------------------------------------------------------------------------
__global__ __launch_bounds__(256) void gni_reduce(const float* __restrict__ x,
                                                  float* __restrict__ sums) {
  __shared__ float tilebuf[2][TILE];   // 32 KB double buffer
  __shared__ float redbuf[16];

  const int b     = blockIdx.x / SPLIT;
  const int chunk = blockIdx.x % SPLIT;
  const float* base =
      x + (size_t)b * N_PER_BATCH + (size_t)chunk * (TILES_PER_BLK * TILE);

  float s = 0.0f, q = 0.0f;

#if defined(HAVE_TDM)
  if (threadIdx.x == 0) {
    tdm_load_tile((uint32_t)(uintptr_t)&tilebuf[0][0], base);
    __builtin_amdgcn_s_wait_tensorcnt(0);
  }
  __syncthreads();

  for (int t = 0; t < TILES_PER_BLK; ++t) {
    if (threadIdx.x == 0 && t + 1 < TILES_PER_BLK)
      tdm_load_tile((uint32_t)(uintptr_t)&tilebuf[(t + 1) & 1][0],
                    base + (size_t)(t + 1) * TILE);

    const float4* p = (const float4*)&tilebuf[t & 1][0];
#pragma unroll
    for (int i = 0; i < (TILE / 4) / 256; ++i) {   // 4 x ds_load_b128 per lane
      float4 v = p[threadIdx.x + i * 256];
      s += v.x + v.y + v.z + v.w;
      q += v.x * v.x + v.y * v.y + v.z * v.z + v.w * v.w;
    }
    __syncthreads();                                // done reading tilebuf[t&1]
    if (threadIdx.x == 0 && t + 1 < TILES_PER_BLK)
      __builtin_amdgcn_s_wait_tensorcnt(0);         // tile t+1 resident
    __syncthreads();
  }
#else
  const float4* p = (const float4*)base;
  for (int i = threadIdx.x; i < TILES_PER_BLK * TILE / 4; i += 256) {
    float4 v = p[i];
    s += v.x + v.y + v.z + v.w;
    q += v.x * v.x + v.y * v.y + v.z * v.z + v.w * v.w;
  }
#endif

  const int lane = threadIdx.x & 31;
  const int wid  = threadIdx.x >> 5;

#if defined(HAVE_WMMA_F32X4)
  // A (16x4): lane l -> A[l&15][0|2] = s, A[l&15][1|3] = q  (VGPR0=s, VGPR1=q).
  // B (4x16): cols 0-7 select even K (s), cols 8-15 select odd K (q).
  // D[m][n<8] = s[m]+s[m+16],  D[m][n>=8] = q[m]+q[m+16].
  v2f a;    a.x = s;                            a.y = q;
  v2f bsel; bsel.x = (lane & 8) ? 0.0f : 1.0f;  bsel.y = (lane & 8) ? 1.0f : 0.0f;
  v8f cz = { 0.0f, 0.0f, 0.0f, 0.0f, 0.0f, 0.0f, 0.0f, 0.0f };
  v8f d = __builtin_amdgcn_wmma_f32_16x16x4_f32(
      /*neg_a=*/false, a, /*neg_b=*/false, bsel,
      /*c_mod=*/(short)0, cz, /*reuse_a=*/false, /*reuse_b=*/false);
  // Each lane holds 8 rows of its half; fold rows, then fold the two halves.
  float local = ((d[0] + d[1]) + (d[2] + d[3])) + ((d[4] + d[5]) + (d[6] + d[7]));
  local += __shfl_xor(local, 16, 32);
  // lanes 0-7: wave sum(s); lanes 8-15: wave sum(q).
  if (lane == 0) redbuf[wid] = local;
  if (lane == 8) redbuf[8 + wid] = local;
#else
#pragma unroll
  for (int off = 16; off > 0; off >>= 1) {
    s += __shfl_down(s, off, 32);
    q += __shfl_down(q, off, 32);
  }
  if (lane == 0) { redbuf[wid] = s; redbuf[8 + wid] = q; }
#endif

  __syncthreads();
  if (threadIdx.x == 0) {
    float ts = 0.0f, tq = 0.0f;
#pragma unroll
    for (int w = 0; w < 8; ++w) { ts += redbuf[w]; tq += redbuf[8 + w]; }
    atomicAdd(&sums[2 * b + 0], ts);
    atomicAdd(&sums[2 * b + 1], tq);
  }
}

// ---------------------------------------------------------------------------
// Pass 2: per-batch quantized stats + integer LUT isqrt (one lane per batch).
// ---------------------------------------------------------------------------
__global__ void gni_stats(const float* __restrict__ sums,
                          const int* __restrict__ lut,
                          float2* __restrict__ stats) {
  const int b = threadIdx.x;
  if (b >= NBATCH) return;

  const float n    = (float)N_PER_BATCH;
  const float sum  = sums[2 * b + 0];
  const float sum2 = sums[2 * b + 1];

  // mu = floor_q(mean, 8, 4)
  float mu = floorf((sum / n) * 16.0f);
  mu = fminf(fmaxf(mu, -128.0f), 127.0f) * 0.0625f;

  // sum((x-mu)^2) = sum(x^2) - 2*mu*sum(x) + n*mu^2 ; floor_q(.,32,8)
  float ssd = sum2 - 2.0f * mu * sum + n * mu * mu;
  ssd = floorf(ssd * 256.0f);
  ssd = fminf(fmaxf(ssd, -2.14748352e9f), 2.14748352e9f) * (1.0f / 256.0f);

  float var = floorf((ssd / n) * 256.0f);
  var = fminf(fmaxf(var, -2.14748352e9f), 2.14748352e9f) * (1.0f / 256.0f);
  var = fminf(fmaxf(var, 0.0f), 65535.0f / 256.0f);
  int v = (int)(var * 256.0f);                       // unsigned 16-bit encoding

  int res;
  if (v == 0) {
    res = 65535;                                     // isqrt(0) saturates
  } else {
    const int msb   = 31 - __clz((unsigned)v);       // MSB index, v in [1,65535]
    const int shift = 15 - msb;
    const unsigned xnorm = (unsigned)v << shift;     // MSB at bit 15
    const int idx = (int)((xnorm >> (15 - 5)) & 31); // top 5 mantissa bits
    int y = lut[idx];                                // ~ 2^15 / sqrt(mant)
    const int e = msb - 8;                           // var = mant * 2^e
    if (e & 1) y = (y * SQRT2_INV_Q15) >> 15;        // odd exponent: * 1/sqrt(2)
    const int ts = (8 - 15) - (e >> 1);              // result scale shift
    res = (ts >= 0) ? (y << ts) : (y >> (-ts));
    res = min(max(res, 0), 65535);
  }
  stats[b] = make_float2(mu, (float)res * (1.0f / 256.0f));
}

// ---------------------------------------------------------------------------
// Pass 3: normalize + Q4 floor cast.  TDM-staged reads (x is L2-resident,
// 103 MB < 192 MB), plain global_store_b128 for the HBM-bound writes.
// Grid: NBATCH * SPLIT blocks of 256 threads, 7 tiles each.
// ---------------------------------------------------------------------------
__global__ __launch_bounds__(256) void gni_norm(const float* __restrict__ x,
                                                const float2* __restrict__ stats,
                                                float* __restrict__ outf,
                                                int* __restrict__ outi) {
  const int b     = blockIdx.x / SPLIT;
  const int chunk = blockIdx.x % SPLIT;
  const size_t base_elem =
      (size_t)b * N_PER_BATCH + (size_t)chunk * (TILES_PER_BLK * TILE);
  const float2 st = stats[b];

#if defined(HAVE_TDM)
  __shared__ float tilebuf[2][TILE];   // 32 KB double buffer

  if (threadIdx.x == 0) {
    tdm_load_tile((uint32_t)(uintptr_t)&tilebuf[0][0], x + base_elem);
    __builtin_amdgcn_s_wait_tensorcnt(0);
  }
  __syncthreads();

  for (int t = 0; t < TILES_PER_BLK; ++t) {
    if (threadIdx.x == 0 && t + 1 < TILES_PER_BLK)
      tdm_load_tile((uint32_t)(uintptr_t)&tilebuf[(t + 1) & 1][0],
                    x + base_elem + (size_t)(t + 1) * TILE);

    const float4* p = (const float4*)&tilebuf[t & 1][0];
    const size_t out4 = (base_elem + (size_t)t * TILE) / 4;
#pragma unroll
    for (int i = 0; i < (TILE / 4) / 256; ++i) {
      const int li = threadIdx.x + i * 256;
      float4 v = p[li];
      float c[4] = { v.x, v.y, v.z, v.w };
      int oi[4]; float of[4];
#pragma unroll
      for (int k = 0; k < 4; ++k) {
        float f = floorf((c[k] - st.x) * st.y * 16.0f);
        f = fminf(fmaxf(f, -128.0f), 127.0f);
        oi[k] = (int)f;
        of[k] = f * 0.0625f;
      }
      ((float4*)outf)[out4 + li] = make_float4(of[0], of[1], of[2], of[3]);
      ((int4*)outi)[out4 + li]   = make_int4(oi[0], oi[1], oi[2], oi[3]);
    }
    __syncthreads();
    if (threadIdx.x == 0 && t + 1 < TILES_PER_BLK)
      __builtin_amdgcn_s_wait_tensorcnt(0);
    __syncthreads();
  }
#else
  const float4* xp = (const float4*)(x + base_elem);
  for (int li = threadIdx.x; li < TILES_PER_BLK * TILE / 4; li += 256) {
    __builtin_prefetch(xp + li + 1024, 0, 0);
    float4 v = xp[li];
    float c[4] = { v.x, v.y, v.z, v.w };
    int oi[4]; float of[4];
#pragma unroll
    for (int k = 0; k < 4; ++k) {
      float f = floorf((c[k] - st.x) * st.y * 16.0f);
      f = fminf(fmaxf(f, -128.0f), 127.0f);
      oi[k] = (int)f;
      of[k] = f * 0.0625f;
    }
    const size_t o4 = base_elem / 4 + li;
    ((float4*)outf)[o4] = make_float4(of[0], of[1], of[2], of[3]);
    ((int4*)outi)[o4]   = make_int4(oi[0], oi[1], oi[2], oi[3]);
  }
#endif
}

// ---------------------------------------------------------------------------
extern "C" void kernel_launch(void* const* d_in, const int* in_sizes, int n_in,
                              void* d_out, int out_size, void* d_ws, size_t ws_size,
                              hipStream_t stream) {
  (void)in_sizes; (void)n_in; (void)out_size; (void)ws_size;

  const float* x   = (const float*)d_in[0];
  const int*   lut = (const int*)d_in[1];

  float*  ws    = (float*)d_ws;
  float*  sums  = ws;                         // 128 * {sum, sumsq}
  float2* stats = (float2*)(ws + 2 * NBATCH); // 128 * {mu, inv_sqrt}

  float* outf = (float*)d_out;                // out_float: TOTAL_ELEMS fp32
  int*   outi = (int*)(outf + TOTAL_ELEMS);   // out_int:   TOTAL_ELEMS int32

  gni_zero<<<1, 256, 0, stream>>>(ws);
  gni_reduce<<<NBATCH * SPLIT, 256, 0, stream>>>(x, sums);
  gni_stats<<<1, NBATCH, 0, stream>>>(sums, lut, stats);
  gni_norm<<<NBATCH * SPLIT, 256, 0, stream>>>(x, stats, outf, outi);
}